// GCNLayer_PyG_4406636446287
// MI455X (gfx1250) — compile-verified
//
#include <hip/hip_runtime.h>
#include <math.h>

// ---------------------------------------------------------------------------
// GCN layer (PyG-style) for MI455X / gfx1250, wave32.
//   1) bn_partial / bn_finalize : deterministic two-pass BatchNorm stats,
//      folded into per-feature affine (scale, shift).
//   2) fused_gemm : Hn = H*scale+shift (fused at LDS load), then
//      Ht = Hn@Wt+bt and Out = Hn@Wo+bo via V_WMMA_F32_16X16X4_F32.
//   3) edge_agg   : per-node (wave per node) sigmoid(dot) scores, softmax
//      over K=17, weighted gather-sum, LeakyReLU. L2-resident gathers.
// ---------------------------------------------------------------------------

#define N_NODES      50000
#define KNB          17
#define DIM          128
#define BN_EPS       1e-5f
#define LRELU_SLOPE  0.01f
#define STATS_BLOCKS 250          // 200 rows per block
#define LDS_STRIDE   130          // 128 + 2 pad: conflict-free, 8B-aligned f2

typedef float v2f __attribute__((ext_vector_type(2)));
typedef float v8f __attribute__((ext_vector_type(8)));

// ---------------- BatchNorm: per-block partial sums (deterministic) --------
__global__ __launch_bounds__(DIM)
void bn_partial_kernel(const float* __restrict__ H, float* __restrict__ partials) {
    const int f = threadIdx.x;                      // feature 0..127
    const int rows = N_NODES / STATS_BLOCKS;        // 200
    const int r0 = blockIdx.x * rows;
    float s = 0.f, ss = 0.f;
    const float* p = H + (size_t)r0 * DIM + f;
    for (int i = 0; i < rows; ++i) {
        float v = p[(size_t)i * DIM];               // coalesced across lanes
        s += v;
        ss += v * v;
    }
    partials[blockIdx.x * (2 * DIM) + f]        = s;
    partials[blockIdx.x * (2 * DIM) + DIM + f]  = ss;
}

// ---------------- BatchNorm: finalize into affine scale/shift --------------
__global__ __launch_bounds__(DIM)
void bn_finalize_kernel(const float* __restrict__ partials,
                        const float* __restrict__ gamma,
                        const float* __restrict__ beta,
                        float* __restrict__ scale,
                        float* __restrict__ shift) {
    const int f = threadIdx.x;
    float s = 0.f, ss = 0.f;
    for (int b = 0; b < STATS_BLOCKS; ++b) {        // fixed order -> bitwise stable
        s  += partials[b * (2 * DIM) + f];
        ss += partials[b * (2 * DIM) + DIM + f];
    }
    const float invN = 1.0f / (float)N_NODES;
    float mean = s * invN;
    float var  = ss * invN - mean * mean;           // biased var (matches jnp.var)
    float sc   = gamma[f] * rsqrtf(var + BN_EPS);
    scale[f] = sc;
    shift[f] = beta[f] - mean * sc;
}

// ---------------- Fused BN + dual GEMM via f32 WMMA ------------------------
// Block: 512 threads = 16 waves, one 16-row slab of Hn.
// Wave w: matrix = w>>3 (0:Wt->Ht, 1:Wo->Out), column tile = (w&7)*16.
__global__ __launch_bounds__(512)
void fused_gemm_kernel(const float* __restrict__ H,
                       const float* __restrict__ scale,
                       const float* __restrict__ shift,
                       const float* __restrict__ Wt, const float* __restrict__ bt,
                       const float* __restrict__ Wo, const float* __restrict__ bo,
                       float* __restrict__ Ht, float* __restrict__ Out) {
    __shared__ float tile[16 * LDS_STRIDE];
    const int tid = threadIdx.x;
    const int rowbase = blockIdx.x * 16;

    // Load 16x128 H tile, apply BN affine, stage in LDS (row stride 130).
    {
        const int r = tid >> 5;                     // 0..15
        const int c = (tid & 31) * 4;               // 0..124
        const float4 hv  = *(const float4*)(H + (size_t)(rowbase + r) * DIM + c);
        const float4 sc4 = *(const float4*)(scale + c);
        const float4 sh4 = *(const float4*)(shift + c);
        float2 lo, hi;
        lo.x = fmaf(hv.x, sc4.x, sh4.x);
        lo.y = fmaf(hv.y, sc4.y, sh4.y);
        hi.x = fmaf(hv.z, sc4.z, sh4.z);
        hi.y = fmaf(hv.w, sc4.w, sh4.w);
        *(float2*)&tile[r * LDS_STRIDE + c]     = lo;   // 8B aligned
        *(float2*)&tile[r * LDS_STRIDE + c + 2] = hi;
    }
    __syncthreads();

    const int wave = tid >> 5;                      // 0..15
    const int lane = tid & 31;
    const int half = lane >> 4;                     // 0 / 1
    const int l16  = lane & 15;

    const float* W   = (wave & 8) ? Wo : Wt;
    const float* Bv  = (wave & 8) ? bo : bt;
    float*       Dst = (wave & 8) ? Out : Ht;
    const int colbase = (wave & 7) * 16;

    // Per-lane bases so the unrolled loop folds into immediate offsets.
    const float* wp  = W + (size_t)(2 * half) * DIM + colbase + l16; // K=ka row
    const float* ap  = &tile[l16 * LDS_STRIDE + 2 * half];

    v8f acc = {};
    #pragma unroll
    for (int k0 = 0; k0 < DIM; k0 += 4) {
        // A fragment (16x4): lane l16 = row M, VGPR0/1 = K=ka, ka+1
        v2f a;
        {
            const float2 av = *(const float2*)(ap + k0);
            a[0] = av.x;
            a[1] = av.y;
        }
        // B fragment (4x16): lane l16 = col N, VGPR0/1 = K=ka, ka+1
        v2f b;
        b[0] = wp[(size_t)k0 * DIM];
        b[1] = wp[(size_t)k0 * DIM + DIM];
        // D = A x B + C   (full f32 precision)
        acc = __builtin_amdgcn_wmma_f32_16x16x4_f32(
            /*neg_a=*/false, a, /*neg_b=*/false, b,
            /*c_mod=*/(short)0, acc, /*reuse_a=*/false, /*reuse_b=*/false);
    }

    // C/D layout: VGPR r -> (M = r + 8*half, N = l16); add bias, store.
    const float bias = Bv[colbase + l16];
    float* dp = Dst + (size_t)(rowbase + 8 * half) * DIM + colbase + l16;
    #pragma unroll
    for (int r = 0; r < 8; ++r) {
        dp[(size_t)r * DIM] = acc[r] + bias;
    }
}

// ---------------- Edge scores + softmax + aggregation ----------------------
// One wave32 per node; lane l owns features 4l..4l+3 (float4).
__global__ __launch_bounds__(256)
void edge_agg_kernel(const float* __restrict__ Ht,
                     const float* __restrict__ Out,
                     const int* __restrict__ col,
                     float* __restrict__ out_f,
                     float* __restrict__ A_out) {
    const int lane = threadIdx.x & 31;
    const int node = blockIdx.x * 8 + (threadIdx.x >> 5);
    if (node >= N_NODES) return;

    const float4* ht4 = (const float4*)Ht;
    const float4* ot4 = (const float4*)Out;

    const float4 s = ht4[(size_t)node * 32 + lane];     // self theta row
    const int* ce = col + (size_t)node * KNB;

    int   cols[KNB];
    float ew[KNB];

    #pragma unroll
    for (int e = 0; e < KNB; ++e) {
        const int cc = ce[e];
        cols[e] = cc;
        const float4 v = ht4[(size_t)cc * 32 + lane];   // coalesced 512B gather
        float p = v.x * s.x + v.y * s.y + v.z * s.z + v.w * s.w;
        p += __shfl_xor(p, 16, 32);                     // wave32 butterfly sum
        p += __shfl_xor(p,  8, 32);
        p += __shfl_xor(p,  4, 32);
        p += __shfl_xor(p,  2, 32);
        p += __shfl_xor(p,  1, 32);
        const float sg = 1.0f / (1.0f + __expf(-p));    // sigmoid, in (0,1)
        ew[e] = __expf(sg);                             // softmax numerator (stable)
    }

    float denom = 0.f;
    #pragma unroll
    for (int e = 0; e < KNB; ++e) denom += ew[e];
    const float inv = 1.0f / denom;

    // Aggregate; also pick up this lane's softmax weight with selects
    // (avoids a dynamic register-array index -> no scratch spill).
    float aval = 0.f;
    float4 acc = {0.f, 0.f, 0.f, 0.f};
    #pragma unroll
    for (int e = 0; e < KNB; ++e) {
        const float a = ew[e] * inv;
        aval = (lane == e) ? a : aval;
        const float4 v = ot4[(size_t)cols[e] * 32 + lane];
        acc.x = fmaf(a, v.x, acc.x);
        acc.y = fmaf(a, v.y, acc.y);
        acc.z = fmaf(a, v.z, acc.z);
        acc.w = fmaf(a, v.w, acc.w);
    }
    acc.x = acc.x >= 0.f ? acc.x : LRELU_SLOPE * acc.x;
    acc.y = acc.y >= 0.f ? acc.y : LRELU_SLOPE * acc.y;
    acc.z = acc.z >= 0.f ? acc.z : LRELU_SLOPE * acc.z;
    acc.w = acc.w >= 0.f ? acc.w : LRELU_SLOPE * acc.w;

    ((float4*)out_f)[(size_t)node * 32 + lane] = acc;
    if (lane < KNB) A_out[(size_t)node * KNB + lane] = aval;
}

// ---------------------------------------------------------------------------
extern "C" void kernel_launch(void* const* d_in, const int* in_sizes, int n_in,
                              void* d_out, int out_size, void* d_ws, size_t ws_size,
                              hipStream_t stream) {
    (void)in_sizes; (void)n_in; (void)out_size; (void)ws_size;

    const float* H     = (const float*)d_in[0];
    const int*   col   = (const int*)d_in[1];
    /* d_in[2] = row: implicit (row[e] = e / K), unused */
    const float* gamma = (const float*)d_in[3];
    const float* beta  = (const float*)d_in[4];
    const float* Wt    = (const float*)d_in[5];
    const float* bt    = (const float*)d_in[6];
    const float* Wo    = (const float*)d_in[7];
    const float* bo    = (const float*)d_in[8];

    float* out_f = (float*)d_out;                        // [N,128]
    float* A_out = out_f + (size_t)N_NODES * DIM;        // [N,17]

    // Workspace layout (floats): scale[128] shift[128] partials[250*256]
    //                            Ht[N*128] Out[N*128]  (~51.5 MB total)
    float* ws       = (float*)d_ws;
    float* scale    = ws;
    float* shift    = ws + DIM;
    float* partials = ws + 2 * DIM;
    float* Ht       = ws + 2 * DIM + STATS_BLOCKS * 2 * DIM;
    float* OutM     = Ht + (size_t)N_NODES * DIM;

    bn_partial_kernel<<<STATS_BLOCKS, DIM, 0, stream>>>(H, partials);
    bn_finalize_kernel<<<1, DIM, 0, stream>>>(partials, gamma, beta, scale, shift);
    fused_gemm_kernel<<<N_NODES / 16, 512, 0, stream>>>(H, scale, shift,
                                                        Wt, bt, Wo, bo, Ht, OutM);
    edge_agg_kernel<<<(N_NODES + 7) / 8, 256, 0, stream>>>(Ht, OutM, col,
                                                           out_f, A_out);
}